// Transformer_26225070310027
// MI455X (gfx1250) — compile-verified
//
#include <hip/hip_runtime.h>

typedef __attribute__((ext_vector_type(2))) float v2f;
typedef __attribute__((ext_vector_type(8))) float v8f;

#define HEADS 8
#define DKDIM 64

// ---------------------------------------------------------------------------
// Zero the workspace accumulators (wv [N*H*DK] and z [N*H], contiguous).
// ---------------------------------------------------------------------------
__global__ void ws_zero_kernel(float* __restrict__ ws, int n) {
    int i = blockIdx.x * blockDim.x + threadIdx.x;
    if (i < n) ws[i] = 0.0f;
}

// ---------------------------------------------------------------------------
// Edge kernel: one wave32 per tile of 16 edges.
//   S(16x16) = Kg(16x64) x Qg(64x16) via 16 chained V_WMMA_F32_16X16X4_F32;
//   diag(S) = per-edge dots. Then s = exp(clamp(dot/8)), scatter-atomic into
//   wv/z (L2-resident).
// A-matrix 16x4 f32 layout: lanes 0-15 -> M=lane, K={0,1}; lanes 16-31 ->
// M=lane-16, K={2,3}. B-matrix 4x16 mirrors it with N=lane&15. So both
// gathers share addressing: element [row][koff + t*4 + {0,1}].
// ---------------------------------------------------------------------------
__global__ void __launch_bounds__(256) edge_attn_scatter_kernel(
    const float* __restrict__ q, const float* __restrict__ k,
    const float* __restrict__ v,
    const int* __restrict__ esrc, const int* __restrict__ edst,
    float* __restrict__ wv, float* __restrict__ z, int E)
{
    __shared__ int   s_src[8][16];
    __shared__ int   s_dst[8][16];
    __shared__ float s_sc [8][16];

    const int lane  = threadIdx.x & 31;
    const int wid   = threadIdx.x >> 5;
    const int ebase = (blockIdx.x * 8 + wid) * 16;
    if (ebase >= E) return;                       // wave-uniform guard

    if (lane < 16) {
        s_src[wid][lane] = esrc[ebase + lane];
        s_dst[wid][lane] = edst[ebase + lane];
    }
    // wave-local LDS; compiler inserts s_wait_dscnt before dependent reads

    const int  mn   = lane & 15;                  // M for A-lanes, N for B-lanes
    const bool hiK  = (lane >= 16);               // upper K pair
    const int  koff = hiK ? 2 : 0;

    const float* krow = k + (size_t)s_src[wid][mn] * (HEADS * DKDIM);
    const float* qrow = q + (size_t)s_dst[wid][mn] * (HEADS * DKDIM);

    for (int h = 0; h < HEADS; ++h) {
        const float* ka = krow + h * DKDIM + koff;
        const float* qb = qrow + h * DKDIM + koff;

        v8f c = {};
        #pragma unroll
        for (int t = 0; t < 16; ++t) {            // K = 64 in steps of 4
            v2f a = *(const v2f*)(ka + t * 4);
            v2f b = *(const v2f*)(qb + t * 4);
            c = __builtin_amdgcn_wmma_f32_16x16x4_f32(
                    false, a, false, b, (short)0, c, false, false);
        }

        // Extract diagonal: (e,e) lives in VGPR e&7 of lane e (e<8) or
        // lane e+16 (e>=8). C/D layout: VGPR r, lane l -> M=r+8*(l>=16), N=l&15.
        float diag = 0.0f;
        #pragma unroll
        for (int r = 0; r < 8; ++r) {
            if ((r + (hiK ? 8 : 0)) == mn) diag = c[r];
        }
        if (hiK == (mn >= 8)) {                   // this lane owns diag(mn)
            float x = diag * 0.125f;              // / sqrt(64)
            x = fminf(10.0f, fmaxf(-10.0f, x));
            s_sc[wid][mn] = __expf(x);
        }

        // z[dst][h] += s  (one lane per edge)
        if (lane < 16) {
            atomicAdd(&z[s_dst[wid][lane] * HEADS + h], s_sc[wid][lane]);
        }

        // wv[dst][h][:] += s * v[src][h][:]  (float2 per lane -> 2 atomics)
        #pragma unroll 4
        for (int e = 0; e < 16; ++e) {
            const float  s  = s_sc[wid][e];
            const float2 ve = *(const float2*)(
                v + ((size_t)s_src[wid][e] * HEADS + h) * DKDIM + lane * 2);
            float* outp = wv + ((size_t)s_dst[wid][e] * HEADS + h) * DKDIM + lane * 2;
            atomicAdd(outp + 0, s * ve.x);
            atomicAdd(outp + 1, s * ve.y);
        }
    }
}

// ---------------------------------------------------------------------------
// out[n,h,d] = wv[n,h,d] / z[n,h]
// ---------------------------------------------------------------------------
__global__ void normalize_kernel(const float* __restrict__ wv,
                                 const float* __restrict__ z,
                                 float* __restrict__ out, int total)
{
    int i = blockIdx.x * blockDim.x + threadIdx.x;
    if (i < total) {
        out[i] = wv[i] / z[i / DKDIM];
    }
}

extern "C" void kernel_launch(void* const* d_in, const int* in_sizes, int n_in,
                              void* d_out, int out_size, void* d_ws, size_t ws_size,
                              hipStream_t stream)
{
    const float* q    = (const float*)d_in[0];
    const float* k    = (const float*)d_in[1];
    const float* v    = (const float*)d_in[2];
    const int*   esrc = (const int*)d_in[3];
    const int*   edst = (const int*)d_in[4];

    const int E   = in_sizes[3];                  // edges
    const int NHD = in_sizes[0];                  // N*H*DK
    const int Nn  = NHD / (HEADS * DKDIM);        // nodes
    const int ZN  = Nn * HEADS;

    float* wv = (float*)d_ws;                     // [N*H*DK]
    float* z  = wv + NHD;                         // [N*H]

    const int totWs = NHD + ZN;
    ws_zero_kernel<<<(totWs + 255) / 256, 256, 0, stream>>>(wv, totWs);

    const int tiles  = (E + 15) / 16;             // 16 edges per wave
    const int blocks = (tiles + 7) / 8;           // 8 waves per block
    edge_attn_scatter_kernel<<<blocks, 256, 0, stream>>>(q, k, v, esrc, edst,
                                                         wv, z, E);

    normalize_kernel<<<(NHD + 255) / 256, 256, 0, stream>>>(wv, z,
                                                            (float*)d_out, NHD);
}